// ResGCNencoder_64364379898083
// MI455X (gfx1250) — compile-verified
//
#include <hip/hip_runtime.h>
#include <hip/hip_bf16.h>

// ---------------------------------------------------------------------------
// ResGCN encoder for MI455X (gfx1250).
//  x:[N,128], edge_idx:[2,E], w1:[128,128], b1:[128], w2:[128,64], b2:[64],
//  wd:[128,64], bd:[64]  ->  out:[N,64]
// Dense GEMMs use V_WMMA_F32_16X16X4_F32 (full f32 precision, wave32).
// Message passing: one wave32 per edge, float4/float2 per lane,
// global_atomic_add_f32 scatter (intermediates live in the 192MB L2).
// ---------------------------------------------------------------------------

typedef __attribute__((ext_vector_type(2))) float v2f;
typedef __attribute__((ext_vector_type(8))) float v8f;

#define NEG_SLOPE 0.01f

__device__ __forceinline__ float leaky(float v) {
    return v > 0.0f ? v : NEG_SLOPE * v;
}

// ---------------- degree / norm --------------------------------------------
__global__ void init_deg_kernel(float* __restrict__ deg, int n) {
    int i = blockIdx.x * blockDim.x + threadIdx.x;
    if (i < n) deg[i] = 1.0f;   // self-loop contributes 1 to every node
}

__global__ void accum_deg_kernel(const int* __restrict__ dst, float* __restrict__ deg, int e) {
    int i = blockIdx.x * blockDim.x + threadIdx.x;
    if (i < e) atomicAdd(&deg[dst[i]], 1.0f);
}

__global__ void dinv_kernel(float* __restrict__ deg, int n) {
    int i = blockIdx.x * blockDim.x + threadIdx.x;
    if (i < n) deg[i] = rsqrtf(deg[i]);   // deg >= 1 always
}

__global__ void zero_f4_kernel(float4* __restrict__ p, long n4) {
    long i = (long)blockIdx.x * blockDim.x + threadIdx.x;
    if (i < n4) p[i] = make_float4(0.f, 0.f, 0.f, 0.f);
}

// ---------------- WMMA GEMM: C = A[M,128] @ W[128,NCOLS] (+bias) ------------
template <int NCOLS>
__global__ void __launch_bounds__(256)
wmma_gemm_kernel(const float* __restrict__ A,     // [M,128] row-major
                 const float* __restrict__ W,     // [128,NCOLS] row-major
                 const float* __restrict__ bias,  // [NCOLS] or nullptr
                 float* __restrict__ C,           // [M,NCOLS]
                 int M) {
    constexpr int K = 128;
    constexpr int ROWS = 32;             // rows per block (2 row-tiles of 16)
    constexpr int LDA = K + 4;           // LDS pad: bank-conflict free
    __shared__ float sA[ROWS * LDA];

    const int tid = threadIdx.x;         // 256 threads = 8 wave32
    const int rowBase = blockIdx.x * ROWS;

    // Cooperative stage of the 32x128 A slab into LDS (float4 per thread).
    for (int i = tid; i < ROWS * (K / 4); i += 256) {
        int r = i / (K / 4);
        int c4 = i % (K / 4);
        int gr = rowBase + r;
        float4 v = make_float4(0.f, 0.f, 0.f, 0.f);
        if (gr < M) v = ((const float4*)(A + (long)gr * K))[c4];
        float* dstp = &sA[r * LDA + c4 * 4];
        dstp[0] = v.x; dstp[1] = v.y; dstp[2] = v.z; dstp[3] = v.w;
    }
    __syncthreads();

    const int wave = tid >> 5;
    const int lane = tid & 31;
    const int half = lane >> 4;          // 0: K=k0..k0+1   1: K=k0+2..k0+3
    const int l16  = lane & 15;

    constexpr int CT = NCOLS / 16;
    constexpr int TILES = 2 * CT;
    for (int t = wave; t < TILES; t += 8) {
        const int rt = t & 1;            // row tile within slab
        const int ct = t >> 1;           // column tile
        const int rowOff = rt * 16;
        const int colBase = ct * 16;

        v8f acc = {};
        const float* sArow = &sA[(rowOff + l16) * LDA];
        #pragma unroll
        for (int k0 = 0; k0 < K; k0 += 4) {
            const int kk = k0 + 2 * half;
            // A frag: 16x4 f32 (lane half picks K pair)
            v2f a;
            a.x = sArow[kk];
            a.y = sArow[kk + 1];
            // B frag: 4x16 f32, N across lanes, K pair per half-wave
            v2f b;
            b.x = W[(long)kk * NCOLS + colBase + l16];
            b.y = W[(long)(kk + 1) * NCOLS + colBase + l16];
            acc = __builtin_amdgcn_wmma_f32_16x16x4_f32(
                false, a, false, b, (short)0, acc, false, false);
        }

        // C/D layout: VGPR i -> M = i (lanes 0-15) / M = 8+i (lanes 16-31)
        const int m0 = rowBase + rowOff + half * 8;
        const int n  = colBase + l16;
        const float bv = bias ? bias[n] : 0.0f;
        #pragma unroll
        for (int i = 0; i < 8; ++i) {
            const int gm = m0 + i;
            if (gm < M) C[(long)gm * NCOLS + n] = acc[i] + bv;
        }
    }
}

// ---------------- edge scatter: acc[dst] += h[src] * dinv[s]*dinv[d] --------
template <int F>
__global__ void __launch_bounds__(256)
scatter_edges_kernel(const int* __restrict__ src, const int* __restrict__ dst,
                     const float* __restrict__ dinv,
                     const float* __restrict__ h,   // [N,F]
                     float* __restrict__ acc,       // [N,F]
                     int E) {
    const int wave = threadIdx.x >> 5;
    const int lane = threadIdx.x & 31;
    const int e = blockIdx.x * (blockDim.x >> 5) + wave;
    if (e >= E) return;

    const int s = src[e];
    const int d = dst[e];
    const float nrm = dinv[s] * dinv[d];

    constexpr int VEC = F / 32;          // 4 (F=128) or 2 (F=64)
    const float* hp = h + (long)s * F + lane * VEC;
    float* ap = acc + (long)d * F + lane * VEC;

    if constexpr (VEC == 4) {
        float4 v = *(const float4*)hp;   // global_load_b128 per lane
        atomicAdd(ap + 0, v.x * nrm);
        atomicAdd(ap + 1, v.y * nrm);
        atomicAdd(ap + 2, v.z * nrm);
        atomicAdd(ap + 3, v.w * nrm);
    } else {
        float2 v = *(const float2*)hp;
        atomicAdd(ap + 0, v.x * nrm);
        atomicAdd(ap + 1, v.y * nrm);
    }
}

// ---------------- finalize: leaky(acc + h*dinv^2 + bias) [+ identity] -------
template <int F>
__global__ void finalize_kernel(const float* __restrict__ acc,
                                const float* __restrict__ h,
                                const float* __restrict__ dinv,
                                const float* __restrict__ bias,
                                const float* __restrict__ identity,  // nullable
                                float* __restrict__ outp,
                                int n) {
    const long i4 = (long)blockIdx.x * blockDim.x + threadIdx.x;
    const long total = (long)n * F / 4;
    if (i4 >= total) return;
    const long base = i4 * 4;
    const int row = (int)(base / F);
    const int col = (int)(base % F);
    const float di = dinv[row];
    const float d2 = di * di;

    float4 a = ((const float4*)acc)[i4];
    float4 hv = ((const float4*)h)[i4];
    float4 r;
    r.x = leaky(a.x + hv.x * d2 + bias[col + 0]);
    r.y = leaky(a.y + hv.y * d2 + bias[col + 1]);
    r.z = leaky(a.z + hv.z * d2 + bias[col + 2]);
    r.w = leaky(a.w + hv.w * d2 + bias[col + 3]);
    if (identity) {
        float4 id = ((const float4*)identity)[i4];
        r.x += id.x; r.y += id.y; r.z += id.z; r.w += id.w;
    }
    ((float4*)outp)[i4] = r;
}

// ---------------------------------------------------------------------------
extern "C" void kernel_launch(void* const* d_in, const int* in_sizes, int n_in,
                              void* d_out, int out_size, void* d_ws, size_t ws_size,
                              hipStream_t stream) {
    const float* x  = (const float*)d_in[0];
    const int* eidx = (const int*)d_in[1];
    const float* w1 = (const float*)d_in[2];
    const float* b1 = (const float*)d_in[3];
    const float* w2 = (const float*)d_in[4];
    const float* b2 = (const float*)d_in[5];
    const float* wd = (const float*)d_in[6];
    const float* bd = (const float*)d_in[7];

    const int N = in_sizes[0] / 128;     // 100000
    const int E = in_sizes[1] / 2;       // 1600000
    const int* src = eidx;               // edge_idx row 0
    const int* dst = eidx + E;           // edge_idx row 1

    // ---- workspace carve-up (floats), 256B aligned regions ----
    float* ws = (float*)d_ws;
    long off = 0;
    auto carve = [&](long cnt) { float* p = ws + off; off += (cnt + 63) & ~63L; return p; };
    float* dinv  = carve(N);             // deg -> dinv in place
    float* h1    = carve((long)N * 128); // x @ w1
    float* acc1  = carve((long)N * 128); // layer-1 aggregate -> out1 (in place)
    float* h2    = carve((long)N * 64);  // out1 @ w2
    float* acc2  = carve((long)N * 64);  // layer-2 aggregate
    float* ident = carve((long)N * 64);  // x @ wd + bd

    const int T = 256;
    // 1) degrees -> dinv
    init_deg_kernel<<<(N + T - 1) / T, T, 0, stream>>>(dinv, N);
    accum_deg_kernel<<<(E + T - 1) / T, T, 0, stream>>>(dst, dinv, E);
    dinv_kernel<<<(N + T - 1) / T, T, 0, stream>>>(dinv, N);

    const int gemmBlocks = (N + 31) / 32;
    // 2) h1 = x @ w1  (bias applied post-aggregation)
    wmma_gemm_kernel<128><<<gemmBlocks, T, 0, stream>>>(x, w1, nullptr, h1, N);
    // 3) identity = x @ wd + bd
    wmma_gemm_kernel<64><<<gemmBlocks, T, 0, stream>>>(x, wd, bd, ident, N);

    // 4) layer-1 message passing
    long n4_1 = (long)N * 128 / 4;
    zero_f4_kernel<<<(unsigned)((n4_1 + T - 1) / T), T, 0, stream>>>((float4*)acc1, n4_1);
    scatter_edges_kernel<128><<<(E + 7) / 8, T, 0, stream>>>(src, dst, dinv, h1, acc1, E);
    finalize_kernel<128><<<(unsigned)((n4_1 + T - 1) / T), T, 0, stream>>>(
        acc1, h1, dinv, b1, nullptr, acc1, N);   // acc1 := out1 (in place)

    // 5) h2 = out1 @ w2
    wmma_gemm_kernel<64><<<gemmBlocks, T, 0, stream>>>(acc1, w2, nullptr, h2, N);

    // 6) layer-2 message passing + residual
    long n4_2 = (long)N * 64 / 4;
    zero_f4_kernel<<<(unsigned)((n4_2 + T - 1) / T), T, 0, stream>>>((float4*)acc2, n4_2);
    scatter_edges_kernel<64><<<(E + 7) / 8, T, 0, stream>>>(src, dst, dinv, h2, acc2, E);
    finalize_kernel<64><<<(unsigned)((n4_2 + T - 1) / T), T, 0, stream>>>(
        acc2, h2, dinv, b2, ident, (float*)d_out, N);
}